// GATGraph_28346784153666
// MI455X (gfx1250) — compile-verified
//
#include <hip/hip_runtime.h>
#include <cstdint>
#include <cstddef>

typedef float v2f __attribute__((ext_vector_type(2)));
typedef float v8f __attribute__((ext_vector_type(8)));

#define IN_CH 128
#define HID 64
#define HEADS 4
#define OUT_CH 2
#define NUM_GRAPHS 64
#define NEG_SLOPE 0.2f

// Try the gfx1250 async global->LDS copy engine (ASYNCcnt path).
// If the inline-asm mnemonic fails to assemble, set to 0 (ds_store fallback).
#define USE_ASYNC_LDS 1

// ---------- ordered-uint encoding for float atomic max ----------
__device__ __forceinline__ unsigned f2key(float f) {
  unsigned u = __float_as_uint(f);
  return (u >> 31) ? ~u : (u | 0x80000000u);
}
__device__ __forceinline__ float key2f(unsigned k) {
  return (k >> 31) ? __uint_as_float(k & 0x7fffffffu) : __uint_as_float(~k);
}
#define NEGINF_KEY 0x007fffffu  // f2key(-inf)

// ---------- fills ----------
__global__ void fill_u32_k(unsigned* __restrict__ p, unsigned v, size_t n) {
  size_t i = (size_t)blockIdx.x * blockDim.x + threadIdx.x;
  if (i < n) p[i] = v;
}

// ---------- fp32 WMMA GEMM ----------
// C[M,NW] = A[M,K] @ B[K,NW].  Block = 256 threads (8 waves).
// Block tile: 128 rows x 64 cols.  B panel (K x 64) staged in LDS, pair-interleaved:
//   ldsB[(k>>1)*128 + n*2 + (k&1)] = B[k][col0+n]   -> lane reads (K=kb,kb+1, N=n) as one b64.
// Each wave: 16 rows x 64 cols via 4 independent f32 16x16x4 WMMA accumulators.
__global__ void gemm_wmma_f32(const float* __restrict__ A, const float* __restrict__ B,
                              float* __restrict__ C, int M, int K, int NW) {
  extern __shared__ float ldsB[];  // K*64 floats
  const int tid = threadIdx.x;
  const int col0 = blockIdx.y << 6;

  // ---- stage B panel into LDS ----
  const int total = K << 6;  // K*64 elements
  for (int idx = tid; idx < total; idx += 256) {
    int k = idx >> 6;
    int n = idx & 63;
    const float* gp = B + (size_t)k * NW + col0 + n;
    int elem = ((k >> 1) << 7) + (n << 1) + (k & 1);
#if USE_ASYNC_LDS
    unsigned la = (unsigned)(uintptr_t)(ldsB + elem);  // low 32 bits == LDS byte offset
    asm volatile("global_load_async_to_lds_b32 %0, %1, off"
                 :: "v"(la), "v"(gp) : "memory");
#else
    ldsB[elem] = *gp;
#endif
  }
#if USE_ASYNC_LDS
  asm volatile("s_wait_asynccnt 0x0" ::: "memory");
#endif
  __syncthreads();

  // ---- compute ----
  const int w = tid >> 5;                       // wave in block: 0..7
  const int mt = blockIdx.x * 8 + w;            // 16-row tile index
  const int mtiles = M >> 4;
  if (mt < mtiles) {
    const int lane = tid & 31;
    const int half = lane >> 4;                 // K-pair select within k-step
    const int l = lane & 15;                    // A row / B,D column within tile
    const float* arow = A + (size_t)((mt << 4) + l) * K;
    v8f acc[4] = {};
    for (int k = 0; k < K; k += 4) {
      int kb = k + (half << 1);
      __builtin_prefetch(arow + kb + 32, 0, 0);
      v2f av = *(const v2f*)(arow + kb);                       // A: M=l, K=kb,kb+1
      const float* bp = ldsB + ((kb >> 1) << 7) + (l << 1);
#pragma unroll
      for (int j = 0; j < 4; ++j) {
        v2f bv = *(const v2f*)(bp + (j << 5));                 // B: K=kb,kb+1, N=l+16j
        acc[j] = __builtin_amdgcn_wmma_f32_16x16x4_f32(false, av, false, bv,
                                                       (short)0, acc[j], false, false);
      }
    }
    // D: vgpr v -> row v + 8*half, col = l (+16j)
    float* cbase = C + (size_t)((mt << 4) + (half << 3)) * NW + col0 + l;
#pragma unroll
    for (int j = 0; j < 4; ++j) {
      float* crow = cbase + (j << 4);
#pragma unroll
      for (int v = 0; v < 8; ++v) crow[(size_t)v * NW] = acc[j][v];
    }
  }
}

// ---------- per-node attention scores ----------
__global__ void attn_scores_k(const float* __restrict__ h, const float* __restrict__ attS,
                              const float* __restrict__ attD, float* __restrict__ aS,
                              float* __restrict__ aD, int N, int H, int C) {
  int i = blockIdx.x * blockDim.x + threadIdx.x;
  if (i >= N * H) return;
  int n = i / H, hh = i - n * H;
  const float* row = h + (size_t)n * H * C + (size_t)hh * C;
  const float* s = attS + hh * C;
  const float* d = attD + hh * C;
  float s1 = 0.f, s2 = 0.f;
  for (int c = 0; c < C; ++c) { float v = row[c]; s1 += v * s[c]; s2 += v * d[c]; }
  aS[i] = s1;
  aD[i] = s2;
}

__device__ __forceinline__ void edge_ends(const int* __restrict__ ei, int e, int E,
                                          int& s, int& d) {
  if (e < E) { s = ei[e]; d = ei[E + e]; }
  else       { s = e - E; d = e - E; }  // self loops appended
}

__device__ __forceinline__ float lrelu(float x) { return x > 0.f ? x : NEG_SLOPE * x; }

// ---------- pass 1: segment max ----------
__global__ void edge_max_k(const int* __restrict__ ei, int E, int E2,
                           const float* __restrict__ aS, const float* __restrict__ aD,
                           unsigned* __restrict__ mkey, int H) {
  int i = blockIdx.x * blockDim.x + threadIdx.x;
  if (i >= E2 * H) return;
  int e = i / H, hh = i - e * H;
  int s, d; edge_ends(ei, e, E, s, d);
  float ev = lrelu(aS[s * H + hh] + aD[d * H + hh]);
  atomicMax(mkey + (size_t)d * H + hh, f2key(ev));
}

// ---------- pass 2: segment sum of exp(e - max) ----------
__global__ void edge_sum_k(const int* __restrict__ ei, int E, int E2,
                           const float* __restrict__ aS, const float* __restrict__ aD,
                           const unsigned* __restrict__ mkey, float* __restrict__ dn, int H) {
  int i = blockIdx.x * blockDim.x + threadIdx.x;
  if (i >= E2 * H) return;
  int e = i / H, hh = i - e * H;
  int s, d; edge_ends(ei, e, E, s, d);
  float ev = lrelu(aS[s * H + hh] + aD[d * H + hh]);
  float ex = expf(ev - key2f(mkey[(size_t)d * H + hh]));
  atomicAdd(dn + (size_t)d * H + hh, ex);
}

// ---------- pass 3: weighted aggregation, thread per 4-channel chunk ----------
__global__ void edge_agg_k(const int* __restrict__ ei, int E, int E2,
                           const float* __restrict__ aS, const float* __restrict__ aD,
                           const unsigned* __restrict__ mkey, const float* __restrict__ dn,
                           const float* __restrict__ h, float* __restrict__ agg,
                           int H, int C) {
  const int F = H * C;
  const int chunks = F >> 2;
  long long i = (long long)blockIdx.x * blockDim.x + threadIdx.x;
  if (i >= (long long)E2 * chunks) return;
  int e = (int)(i / chunks);
  int ci = (int)(i - (long long)e * chunks);
  int c0 = ci << 2;           // 4-chunk never crosses a head boundary (4 | C)
  int hh = c0 / C;
  int s, d; edge_ends(ei, e, E, s, d);
  float ev = lrelu(aS[s * H + hh] + aD[d * H + hh]);
  float ex = expf(ev - key2f(mkey[(size_t)d * H + hh]));
  float alpha = ex / (dn[(size_t)d * H + hh] + 1e-16f);
  const float4 hv = *(const float4*)(h + (size_t)s * F + c0);   // global_load_b128
  float* ap = agg + (size_t)d * F + c0;
  atomicAdd(ap + 0, alpha * hv.x);
  atomicAdd(ap + 1, alpha * hv.y);
  atomicAdd(ap + 2, alpha * hv.z);
  atomicAdd(ap + 3, alpha * hv.w);
}

// ---------- bias + layernorm + ELU, in place, one wave per row ----------
__global__ void bias_ln_elu_k(float* __restrict__ x, const float* __restrict__ bias,
                              const float* __restrict__ g, const float* __restrict__ b,
                              int N, int F) {
  int wpb = blockDim.x >> 5;
  int row = blockIdx.x * wpb + (threadIdx.x >> 5);
  if (row >= N) return;
  int lane = threadIdx.x & 31;
  float* p = x + (size_t)row * F;
  float vals[8];
  float s = 0.f, s2 = 0.f;
  int k = 0;
  for (int c = lane; c < F; c += 32) {
    float v = p[c] + bias[c];
    vals[k++] = v;
    s += v;
    s2 += v * v;
  }
  for (int off = 16; off; off >>= 1) {
    s  += __shfl_xor(s, off, 32);
    s2 += __shfl_xor(s2, off, 32);
  }
  float mu  = s / F;
  float var = s2 / F - mu * mu;
  float r   = rsqrtf(var + 1e-5f);
  k = 0;
  for (int c = lane; c < F; c += 32) {
    float y = (vals[k++] - mu) * r * g[c] + b[c];
    p[c] = y > 0.f ? y : expm1f(y);
  }
}

// ---------- global mean pool (accumulate) ----------
__global__ void pool_k(const float* __restrict__ h, const int* __restrict__ batch,
                       float* __restrict__ sums, float* __restrict__ cnt, int N) {
  int i = blockIdx.x * blockDim.x + threadIdx.x;
  if (i >= N * HID) return;
  int n = i / HID, c = i - n * HID;
  int g = batch[n];
  atomicAdd(sums + (size_t)g * HID + c, h[i]);
  if (c == 0) atomicAdd(cnt + g, 1.0f);
}

// ---------- final FC on [NUM_GRAPHS, HID] ----------
__global__ void final_k(const float* __restrict__ sums, const float* __restrict__ cnt,
                        const float* __restrict__ fcW, const float* __restrict__ fcb,
                        float* __restrict__ out) {
  int t = blockIdx.x * blockDim.x + threadIdx.x;
  if (t >= NUM_GRAPHS * OUT_CH) return;
  int g = t / OUT_CH, o = t - g * OUT_CH;
  float inv = 1.0f / fmaxf(cnt[g], 1.0f);
  float acc = fcb[o];
  for (int c = 0; c < HID; ++c) acc += sums[(size_t)g * HID + c] * inv * fcW[c * OUT_CH + o];
  out[t] = acc;
}

static inline unsigned nblk(size_t n, int bs) { return (unsigned)((n + bs - 1) / bs); }

extern "C" void kernel_launch(void* const* d_in, const int* in_sizes, int n_in,
                              void* d_out, int out_size, void* d_ws, size_t ws_size,
                              hipStream_t stream) {
  const float* x    = (const float*)d_in[0];
  const int*   ei   = (const int*)d_in[1];
  const int*   bat  = (const int*)d_in[2];
  const float* W1   = (const float*)d_in[3];
  const float* atS1 = (const float*)d_in[4];
  const float* atD1 = (const float*)d_in[5];
  const float* b1   = (const float*)d_in[6];
  const float* g1   = (const float*)d_in[7];
  const float* be1  = (const float*)d_in[8];
  const float* W2   = (const float*)d_in[9];
  const float* atS2 = (const float*)d_in[10];
  const float* atD2 = (const float*)d_in[11];
  const float* b2   = (const float*)d_in[12];
  const float* g2   = (const float*)d_in[13];
  const float* be2  = (const float*)d_in[14];
  const float* fcW  = (const float*)d_in[15];
  const float* fcb  = (const float*)d_in[16];
  float* out = (float*)d_out;

  const int N  = in_sizes[0] / IN_CH;  // 50000 (divisible by 16)
  const int E  = in_sizes[1] / 2;      // 800000
  const int E2 = E + N;                // + self loops
  const int F1 = HEADS * HID;          // 256

  float* ws = (float*)d_ws;
  size_t o = 0;
  float*    h1   = ws + o; o += (size_t)N * F1;
  float*    agg1 = ws + o; o += (size_t)N * F1;   // becomes hpost1 in place
  float*    h2   = ws + o; o += (size_t)N * HID;
  float*    agg2 = ws + o; o += (size_t)N * HID;  // becomes hpost2 in place
  float*    aS1  = ws + o; o += (size_t)N * HEADS;
  float*    aD1  = ws + o; o += (size_t)N * HEADS;
  unsigned* m1   = (unsigned*)(ws + o); o += (size_t)N * HEADS;
  float*    dn1  = ws + o; o += (size_t)N * HEADS;
  float*    aS2  = ws + o; o += (size_t)N;
  float*    aD2  = ws + o; o += (size_t)N;
  unsigned* m2   = (unsigned*)(ws + o); o += (size_t)N;
  float*    dn2  = ws + o; o += (size_t)N;
  float*    sums = ws + o; o += (size_t)NUM_GRAPHS * HID;
  float*    cnt  = ws + o; o += (size_t)NUM_GRAPHS;

  // init accumulators
  fill_u32_k<<<nblk((size_t)N * F1, 256), 256, 0, stream>>>((unsigned*)agg1, 0u, (size_t)N * F1);
  fill_u32_k<<<nblk((size_t)N * HID, 256), 256, 0, stream>>>((unsigned*)agg2, 0u, (size_t)N * HID);
  fill_u32_k<<<nblk((size_t)N * HEADS, 256), 256, 0, stream>>>((unsigned*)dn1, 0u, (size_t)N * HEADS);
  fill_u32_k<<<nblk((size_t)N, 256), 256, 0, stream>>>((unsigned*)dn2, 0u, (size_t)N);
  fill_u32_k<<<nblk((size_t)N * HEADS, 256), 256, 0, stream>>>(m1, NEGINF_KEY, (size_t)N * HEADS);
  fill_u32_k<<<nblk((size_t)N, 256), 256, 0, stream>>>(m2, NEGINF_KEY, (size_t)N);
  fill_u32_k<<<nblk((size_t)NUM_GRAPHS * HID, 256), 256, 0, stream>>>((unsigned*)sums, 0u, (size_t)NUM_GRAPHS * HID);
  fill_u32_k<<<1, 64, 0, stream>>>((unsigned*)cnt, 0u, (size_t)NUM_GRAPHS);

  const int mblocks = ((N >> 4) + 7) / 8;  // 8 row-tiles (128 rows) per block

  // ---- layer 1 ----
  gemm_wmma_f32<<<dim3(mblocks, F1 / 64), 256, (size_t)IN_CH * 64 * 4, stream>>>(
      x, W1, h1, N, IN_CH, F1);
  attn_scores_k<<<nblk((size_t)N * HEADS, 256), 256, 0, stream>>>(h1, atS1, atD1, aS1, aD1, N, HEADS, HID);
  edge_max_k<<<nblk((size_t)E2 * HEADS, 256), 256, 0, stream>>>(ei, E, E2, aS1, aD1, m1, HEADS);
  edge_sum_k<<<nblk((size_t)E2 * HEADS, 256), 256, 0, stream>>>(ei, E, E2, aS1, aD1, m1, dn1, HEADS);
  edge_agg_k<<<nblk((size_t)E2 * (F1 >> 2), 256), 256, 0, stream>>>(ei, E, E2, aS1, aD1, m1, dn1, h1, agg1, HEADS, HID);
  bias_ln_elu_k<<<nblk(N, 8), 256, 0, stream>>>(agg1, b1, g1, be1, N, F1);

  // ---- layer 2 ----
  gemm_wmma_f32<<<dim3(mblocks, HID / 64), 256, (size_t)F1 * 64 * 4, stream>>>(
      agg1, W2, h2, N, F1, HID);
  attn_scores_k<<<nblk(N, 256), 256, 0, stream>>>(h2, atS2, atD2, aS2, aD2, N, 1, HID);
  edge_max_k<<<nblk(E2, 256), 256, 0, stream>>>(ei, E, E2, aS2, aD2, m2, 1);
  edge_sum_k<<<nblk(E2, 256), 256, 0, stream>>>(ei, E, E2, aS2, aD2, m2, dn2, 1);
  edge_agg_k<<<nblk((size_t)E2 * (HID >> 2), 256), 256, 0, stream>>>(ei, E, E2, aS2, aD2, m2, dn2, h2, agg2, 1, HID);
  bias_ln_elu_k<<<nblk(N, 8), 256, 0, stream>>>(agg2, b2, g2, be2, N, HID);

  // ---- pool + fc ----
  pool_k<<<nblk((size_t)N * HID, 256), 256, 0, stream>>>(agg2, bat, sums, cnt, N);
  final_k<<<1, 128, 0, stream>>>(sums, cnt, fcW, fcb, out);
}